// NeighborAttention_40922448396352
// MI455X (gfx1250) — compile-verified
//
#include <hip/hip_runtime.h>
#include <hip/hip_bf16.h>
#include <math.h>

// Problem constants (from reference): N=50000, K=1, M=32, D=128.
#define D_DIM 128
#define M_DIM 32
#define WAVES_PER_WG 8
#define THREADS_PER_WG (WAVES_PER_WG * 32)
#define CHUNK_MROWS 8                         // m-rows per async chunk
#define CHUNK_FLOATS (CHUNK_MROWS * D_DIM)    // 1024 floats = 4 KB
#define NCHUNKS (M_DIM / CHUNK_MROWS)         // 4
#define BUF_FLOATS (2 * CHUNK_FLOATS)         // double buffer per wave

// One async b128 instruction copies 32 lanes * 16B = 512B, with the same
// immediate offset applied to BOTH the global and the LDS address
// (CDNA5 ISA).  8 of them move one 4 KB chunk.
#define ASYNC1(OFS)                                                          \
  asm volatile("global_load_async_to_lds_b128 %0, %1, off offset:" OFS       \
               :: "v"(la), "v"(ga) : "memory")
#define ASYNC_CHUNK()                                                        \
  do { ASYNC1("0");    ASYNC1("512");  ASYNC1("1024"); ASYNC1("1536");       \
       ASYNC1("2048"); ASYNC1("2560"); ASYNC1("3072"); ASYNC1("3584"); } while (0)

__global__ void __launch_bounds__(THREADS_PER_WG)
logits_kernel(const float* __restrict__ node1, const float* __restrict__ node2,
              float* __restrict__ logits, int N) {
  // 8 waves * 2 buffers * 4 KB = 64 KB LDS (WGP has 320 KB -> good occupancy)
  __shared__ __align__(16) float smem[WAVES_PER_WG * BUF_FLOATS];

  const int lane = threadIdx.x & 31;
  const int wave = threadIdx.x >> 5;
  const int wavesTotal = gridDim.x * WAVES_PER_WG;

  float* wbuf = &smem[wave * BUF_FLOATS];
  // Generic shared VA: low 32 bits are the LDS byte offset (aperture rule).
  const unsigned int ldsBase =
      (unsigned int)(size_t)wbuf + (unsigned int)(lane * 16);

  int n = blockIdx.x * WAVES_PER_WG + wave;
  for (; n < N; n += wavesTotal) {
    // node1 slice: lane l owns d = 4l..4l+3 (matches node2 flattening below)
    const float4 a4 =
        *reinterpret_cast<const float4*>(node1 + (size_t)n * D_DIM + lane * 4);

    const unsigned long long gbase =
        (unsigned long long)(const char*)(node2 + (size_t)n * (M_DIM * D_DIM)) +
        (unsigned long long)(lane * 16);

    // Prefetch chunk 0 into buffer 0.
    {
      unsigned long long ga = gbase;
      unsigned int la = ldsBase;
      ASYNC_CHUNK();
    }

    float4 acc = make_float4(0.f, 0.f, 0.f, 0.f);
#pragma unroll
    for (int c = 0; c < NCHUNKS; ++c) {
      if (c < NCHUNKS - 1) {
        // Make sure prior DS reads of the buffer we are about to overwrite
        // have been serviced before the async write-back can land.
        asm volatile("s_wait_dscnt 0x0" ::: "memory");
        unsigned long long ga = gbase + (unsigned long long)((c + 1) * 4096);
        unsigned int la = ldsBase + (unsigned int)(((c + 1) & 1) * (CHUNK_FLOATS * 4));
        ASYNC_CHUNK();
        // 8 newer ops in flight; oldest 8 (chunk c) complete in order.
        asm volatile("s_wait_asynccnt 0x8" ::: "memory");
      } else {
        asm volatile("s_wait_asynccnt 0x0" ::: "memory");
      }
      const float* buf = wbuf + (c & 1) * CHUNK_FLOATS + lane * 4;
#pragma unroll
      for (int r = 0; r < CHUNK_MROWS; ++r) {
        float4 v = *reinterpret_cast<const float4*>(buf + r * D_DIM);
        acc.x += v.x; acc.y += v.y; acc.z += v.z; acc.w += v.w;
      }
    }

    // logit[n] = dot(node1[n], sum_m node2[n,m]) ; wave32 butterfly reduction
    float s = acc.x * a4.x + acc.y * a4.y + acc.z * a4.z + acc.w * a4.w;
#pragma unroll
    for (int off = 16; off >= 1; off >>= 1) s += __shfl_xor(s, off, 32);
    if (lane == 0) logits[n] = s;
  }
}

// Single-block reduction over all logits -> {max, sum(exp(x-max))} in stats.
__global__ void __launch_bounds__(1024)
softmax_reduce_kernel(const float* __restrict__ logits, int N,
                      float* __restrict__ stats) {
  __shared__ float red[1024];
  const int tid = threadIdx.x;

  float m = -INFINITY;
  for (int i = tid; i < N; i += 1024) m = fmaxf(m, logits[i]);
  red[tid] = m;
  __syncthreads();
  for (int s = 512; s > 0; s >>= 1) {
    if (tid < s) red[tid] = fmaxf(red[tid], red[tid + s]);
    __syncthreads();
  }
  const float gmax = red[0];
  __syncthreads();

  float sum = 0.f;
  for (int i = tid; i < N; i += 1024) sum += __expf(logits[i] - gmax);
  red[tid] = sum;
  __syncthreads();
  for (int s = 512; s > 0; s >>= 1) {
    if (tid < s) red[tid] += red[tid + s];
    __syncthreads();
  }
  if (tid == 0) {
    stats[0] = gmax;
    stats[1] = red[0];
  }
}

__global__ void softmax_norm_kernel(float* __restrict__ out, int N,
                                    const float* __restrict__ stats) {
  const int i = blockIdx.x * blockDim.x + threadIdx.x;
  if (i < N) {
    const float inv = 1.0f / stats[1];
    out[i] = __expf(out[i] - stats[0]) * inv;
  }
}

extern "C" void kernel_launch(void* const* d_in, const int* in_sizes, int n_in,
                              void* d_out, int out_size, void* d_ws, size_t ws_size,
                              hipStream_t stream) {
  const float* node1 = (const float*)d_in[0];  // [N,1,128]
  const float* node2 = (const float*)d_in[1];  // [N,32,128]
  float* out = (float*)d_out;                  // [N,1] -> logits, then scores
  float* stats = (float*)d_ws;                 // [max, sumexp]

  const int N = in_sizes[0] / D_DIM;  // K=1

  const int blocks = (N + WAVES_PER_WG - 1) / WAVES_PER_WG;  // 1 wave / node
  logits_kernel<<<blocks, THREADS_PER_WG, 0, stream>>>(node1, node2, out, N);
  softmax_reduce_kernel<<<1, 1024, 0, stream>>>(out, N, stats);
  softmax_norm_kernel<<<(N + 255) / 256, 256, 0, stream>>>(out, N, stats);
}